// RGCNLayer_38044820308122
// MI455X (gfx1250) — compile-verified
//
#include <hip/hip_runtime.h>

typedef float v2f __attribute__((ext_vector_type(2)));
typedef float v4f __attribute__((ext_vector_type(4)));
typedef float v8f __attribute__((ext_vector_type(8)));

#define IN_F   64
#define OUT_F  64
#define NBASE  4
#define YCOLS  256   // NBASE * OUT_F

// ------------------------------------------------------------------
// Kernel 1: zero the output accumulator (B128 stores)
// ------------------------------------------------------------------
__global__ void zero_out_kernel(v4f* __restrict__ out, int n4) {
    int i = blockIdx.x * blockDim.x + threadIdx.x;
    if (i < n4) out[i] = (v4f){0.f, 0.f, 0.f, 0.f};
}

// ------------------------------------------------------------------
// Kernel 2: Y[n, b*64+o] = sum_i h[n,i] * weight[b,i,o]
// Dense GEMM [N x 64] @ [64 x 256] using V_WMMA_F32_16X16X4_F32.
// grid = (N/16, 2); block = 256 threads = 8 waves; wave -> 16x16 tile.
// (Verified: lowers to 16x v_wmma_f32_16x16x4_f32 with ds_load_2addr
//  pairs overlapping the matrix pipe.)
// ------------------------------------------------------------------
#define HS_STRIDE 68    // row-to-row bank offset 4 -> conflict-free A-frag reads
#define WS_STRIDE 136   // 2-row offset = 16 banks -> lanes 0-15 vs 16-31 disjoint

__global__ void __launch_bounds__(256)
basis_gemm_kernel(const float* __restrict__ h,       // [N][64]
                  const float* __restrict__ weight,  // [4][64][64]
                  float* __restrict__ Y,             // [N][256]
                  int N)
{
    __shared__ float hs[16 * HS_STRIDE];
    __shared__ float Wl[64 * WS_STRIDE];

    const int tid  = threadIdx.x;
    const int wave = tid >> 5;          // 0..7 (wave32)
    const int lane = tid & 31;

    const int row0  = blockIdx.x * 16;  // node tile base
    const int chalf = blockIdx.y * 128; // which 2 bases (cols 0-127 or 128-255)

    // stage h tile: 16 x 64 (clamp row index -> EXEC stays full, no divergence)
    for (int i = tid; i < 16 * 64; i += 256) {
        int r = i >> 6, c = i & 63;
        int gr = row0 + r; gr = gr < N ? gr : N - 1;
        hs[r * HS_STRIDE + c] = h[(size_t)gr * IN_F + c];
    }
    // stage weight half-panel: 64 x 128.  Global col c -> basis c>>6, out-col c&63.
    for (int i = tid; i < 64 * 128; i += 256) {
        int r = i >> 7, cc = i & 127;
        int c = chalf + cc;
        Wl[r * WS_STRIDE + cc] = weight[((c >> 6) * IN_F + r) * OUT_F + (c & 63)];
    }
    __syncthreads();

    const int cc0 = wave * 16;          // 16-col tile within the 128-wide half
    const int mn  = lane & 15;          // A row / B col for this lane
    const int kh  = (lane >> 4) * 2;    // K sub-offset: lanes 0-15 -> K 0,1 ; 16-31 -> K 2,3

    v8f acc = {};
    #pragma unroll
    for (int k0 = 0; k0 < IN_F; k0 += 4) {
        v2f a, b;
        a.x = hs[mn * HS_STRIDE + k0 + kh];
        a.y = hs[mn * HS_STRIDE + k0 + kh + 1];
        b.x = Wl[(k0 + kh)     * WS_STRIDE + cc0 + mn];
        b.y = Wl[(k0 + kh + 1) * WS_STRIDE + cc0 + mn];
        // 8 args: (neg_a, A, neg_b, B, c_mod, C, reuse_a, reuse_b)
        acc = __builtin_amdgcn_wmma_f32_16x16x4_f32(false, a, false, b,
                                                    (short)0, acc, false, false);
    }

    // C/D layout: VGPR v, lanes 0-15 -> M=v ; lanes 16-31 -> M=v+8 ; N = lane&15
    const int col  = chalf + cc0 + (lane & 15);
    const int mrow = (lane >> 4) * 8;
    #pragma unroll
    for (int v = 0; v < 8; ++v) {
        int r = row0 + mrow + v;
        if (r < N) Y[(size_t)r * YCOLS + col] = acc[v];
    }
}

// ------------------------------------------------------------------
// Kernel 3: per-edge basis combine + scatter-add.
// 16 threads per edge, float4 (B128) gathers: same L2 bytes as before
// but 4x fewer vmem instructions. Y (51.2 MB) is L2-resident so gathers
// and the f32 atomics stay on-chip.
// ------------------------------------------------------------------
__global__ void __launch_bounds__(256)
edge_scatter_kernel(const float* __restrict__ Y,        // [N][256]
                    const float* __restrict__ w_comp,   // [32][4]
                    const float* __restrict__ edge_norm,
                    const int*   __restrict__ src,
                    const int*   __restrict__ dst,
                    const int*   __restrict__ etype,
                    float*       __restrict__ out,      // [N][64]
                    int E)
{
    int t  = blockIdx.x * 256 + threadIdx.x;
    int e  = t >> 4;           // edge index
    int o4 = (t & 15) * 4;     // first of 4 output features handled by this lane
    if (e >= E) return;

    int   s   = src[e];
    int   d   = dst[e];
    int   r   = etype[e];
    float nrm = edge_norm[e];

    // 512 B coefficient table -> cache-resident; fold norm in once.
    const float* cw = w_comp + r * NBASE;
    float c0 = cw[0] * nrm, c1 = cw[1] * nrm, c2 = cw[2] * nrm, c3 = cw[3] * nrm;

    const float* ys = Y + (size_t)s * YCOLS;           // 1 KB row, 16B-aligned slices
    v4f y0 = *(const v4f*)(ys +       o4);
    v4f y1 = *(const v4f*)(ys +  64 + o4);
    v4f y2 = *(const v4f*)(ys + 128 + o4);
    v4f y3 = *(const v4f*)(ys + 192 + o4);

    v4f v = c0 * y0 + c1 * y1 + c2 * y2 + c3 * y3;

    float* op = out + (size_t)d * OUT_F + o4;
    atomicAdd(op + 0, v.x);    // global_atomic_add_f32 @ L2
    atomicAdd(op + 1, v.y);
    atomicAdd(op + 2, v.z);
    atomicAdd(op + 3, v.w);
}

// ------------------------------------------------------------------
// Kernel 4: out = relu(out + bias), B128 in/out
// ------------------------------------------------------------------
__global__ void bias_relu_kernel(v4f* __restrict__ out,
                                 const float* __restrict__ bias, int n4) {
    int i = blockIdx.x * blockDim.x + threadIdx.x;
    if (i < n4) {
        int o4 = (i & 15) * 4;            // 64 floats = 16 vec4 per node row
        v4f b = *(const v4f*)(bias + o4);
        v4f v = out[i] + b;
        v.x = v.x > 0.f ? v.x : 0.f;
        v.y = v.y > 0.f ? v.y : 0.f;
        v.z = v.z > 0.f ? v.z : 0.f;
        v.w = v.w > 0.f ? v.w : 0.f;
        out[i] = v;
    }
}

// ------------------------------------------------------------------
extern "C" void kernel_launch(void* const* d_in, const int* in_sizes, int n_in,
                              void* d_out, int out_size, void* d_ws, size_t ws_size,
                              hipStream_t stream) {
    const float* h         = (const float*)d_in[0];
    const float* weight    = (const float*)d_in[1];
    const float* w_comp    = (const float*)d_in[2];
    const float* bias      = (const float*)d_in[3];
    const float* edge_norm = (const float*)d_in[4];
    const int*   src       = (const int*)  d_in[5];
    const int*   dst       = (const int*)  d_in[6];
    const int*   etype     = (const int*)  d_in[7];

    float* out = (float*)d_out;
    float* Y   = (float*)d_ws;              // [N][256] = 51.2 MB scratch

    const int N    = in_sizes[0] / IN_F;    // 50000
    const int E    = in_sizes[4];           // 800000
    const int nout = out_size;              // N * 64
    const int n4   = nout / 4;

    zero_out_kernel<<<(n4 + 255) / 256, 256, 0, stream>>>((v4f*)out, n4);

    dim3 ggrid((N + 15) / 16, 2);
    basis_gemm_kernel<<<ggrid, 256, 0, stream>>>(h, weight, Y, N);

    long long ethreads = (long long)E * 16;
    edge_scatter_kernel<<<(int)((ethreads + 255) / 256), 256, 0, stream>>>(
        Y, w_comp, edge_norm, src, dst, etype, out, E);

    bias_relu_kernel<<<(n4 + 255) / 256, 256, 0, stream>>>((v4f*)out, bias, n4);
}